// node_centre_encoding_block_55362128445870
// MI455X (gfx1250) — compile-verified
//
#include <hip/hip_runtime.h>
#include <hip/hip_bf16.h>

// ---------------------------------------------------------------------------
// Fused node-centre encoding block for gfx1250 (MI455X), fp32 WMMA path.
//
// Decomposition (C = N = 1024):
//   a_i  = softmax( (Qi @ Ki^T) / 4 )                       [C,N]
//   Sb   = Qe @ Kbase^T ;  Si = Qe @ Kinit^T                [C,N]
//   a_e  = softmax( (Sb + a_i*Si) / 8 )                     [C,N]
//   g    = sigmoid( a_e*(gb + a_i*gi) + bg )                [C,N]
//   out  = (g*a_e) @ Vbase + (g*a_e*a_i) @ Vinit            [C,64]
// where Kbase = X@Wk_e[:6], Kinit = Vi@Wk_e[6:], etc. (all [N,64]).
// ---------------------------------------------------------------------------

typedef float v2f __attribute__((ext_vector_type(2)));
typedef float v8f __attribute__((ext_vector_type(8)));

#define WMMA_F32(A, B, C) \
  __builtin_amdgcn_wmma_f32_16x16x4_f32(false, (A), false, (B), (short)0, (C), false, false)

#define NN 1024
#define LSTR 1028   // LDS row stride (floats); 1028 % 64 == 4 -> conflict-free frag loads

// ---- workspace layout (float offsets) ----
#define QI_OFF   0
#define KI_OFF   (QI_OFF + NN * 16)
#define QE_OFF   (KI_OFF + NN * 16)
#define KB_OFF   (QE_OFF + NN * 64)
#define KIN_OFF  (KB_OFF + NN * 64)
#define VB_OFF   (KIN_OFF + NN * 64)
#define VIN_OFF  (VB_OFF + NN * 64)
#define GB_OFF   (VIN_OFF + NN * 64)
#define GI_OFF   (GB_OFF + NN)
#define WS_FLOATS (GI_OFF + NN)

// ---------------------------------------------------------------------------
// Kernel 1: per-node projections (tiny: 1024 threads, ~3.5K FMA each).
// ---------------------------------------------------------------------------
__global__ __launch_bounds__(256) void nce_precompute(
    const float* __restrict__ X,     // [1024,6]
    const float* __restrict__ Wq_i,  // [6,16]
    const float* __restrict__ Wk_i,  // [6,16]
    const float* __restrict__ Wv_i,  // [6,16]
    const float* __restrict__ Wq_e,  // [6,64]
    const float* __restrict__ Wk_e,  // [22,64]
    const float* __restrict__ Wv_e,  // [22,64]
    const float* __restrict__ wg,    // [64]
    float* __restrict__ ws)
{
  const int n = blockIdx.x * blockDim.x + threadIdx.x;
  if (n >= NN) return;

  float x[6];
#pragma unroll
  for (int f = 0; f < 6; ++f) x[f] = X[n * 6 + f];

  // Vi kept in registers (feeds Kinit/Vinit); Qi/Ki stored.
  float Vi[16];
#pragma unroll
  for (int j = 0; j < 16; ++j) {
    float qv = 0.f, kv = 0.f, vv = 0.f;
#pragma unroll
    for (int f = 0; f < 6; ++f) {
      qv += x[f] * Wq_i[f * 16 + j];
      kv += x[f] * Wk_i[f * 16 + j];
      vv += x[f] * Wv_i[f * 16 + j];
    }
    ws[QI_OFF + n * 16 + j] = qv;
    ws[KI_OFF + n * 16 + j] = kv;
    Vi[j] = vv;
  }

  float gb = 0.f, gi = 0.f;
  for (int j = 0; j < 64; ++j) {
    float qe = 0.f, kb = 0.f, vb = 0.f;
#pragma unroll
    for (int f = 0; f < 6; ++f) {
      qe += x[f] * Wq_e[f * 64 + j];
      kb += x[f] * Wk_e[f * 64 + j];
      vb += x[f] * Wv_e[f * 64 + j];
    }
    float kin = 0.f, vin = 0.f;
#pragma unroll
    for (int t = 0; t < 16; ++t) {
      kin += Vi[t] * Wk_e[(6 + t) * 64 + j];
      vin += Vi[t] * Wv_e[(6 + t) * 64 + j];
    }
    ws[QE_OFF  + n * 64 + j] = qe;
    ws[KB_OFF  + n * 64 + j] = kb;
    ws[KIN_OFF + n * 64 + j] = kin;
    ws[VB_OFF  + n * 64 + j] = vb;
    ws[VIN_OFF + n * 64 + j] = vin;
    float w = wg[j];
    gb += vb * w;
    gi += vin * w;
  }
  ws[GB_OFF + n] = gb;
  ws[GI_OFF + n] = gi;
}

// ---------------------------------------------------------------------------
// Row softmax over a [16 x 1024] LDS buffer (stride LSTR), 256 threads.
// red: [16*17] scratch.
// ---------------------------------------------------------------------------
__device__ __forceinline__ void softmax_rows(float* buf, float* red, int tid) {
  const int row = tid >> 4;   // 0..15
  const int sub = tid & 15;   // 0..15
  float* rp = buf + row * LSTR;

  float m = -3.402823466e38f;
  for (int c = sub; c < NN; c += 16) m = fmaxf(m, rp[c]);
  red[row * 17 + sub] = m;
  __syncthreads();
  if (sub == 0) {
    float mm = red[row * 17];
#pragma unroll
    for (int i = 1; i < 16; ++i) mm = fmaxf(mm, red[row * 17 + i]);
    red[row * 17] = mm;
  }
  __syncthreads();
  const float rowmax = red[row * 17];
  __syncthreads();  // everyone has rowmax before red is overwritten

  float s = 0.f;
  for (int c = sub; c < NN; c += 16) {
    float e = __expf(rp[c] - rowmax);
    rp[c] = e;
    s += e;
  }
  red[row * 17 + sub] = s;
  __syncthreads();
  if (sub == 0) {
    float ss = red[row * 17];
#pragma unroll
    for (int i = 1; i < 16; ++i) ss += red[row * 17 + i];
    red[row * 17] = ss;
  }
  __syncthreads();
  const float inv = 1.0f / red[row * 17];
  for (int c = sub; c < NN; c += 16) rp[c] *= inv;
}

// ---------------------------------------------------------------------------
// Kernel 2: fused attention block. One block = 16 centres, 8 waves.
// ---------------------------------------------------------------------------
__global__ __launch_bounds__(256) void nce_fused(
    const float* __restrict__ ws,
    const float* __restrict__ bgp,
    float* __restrict__ out)
{
  extern __shared__ float smem[];
  float* bufA = smem;                 // [16][LSTR] : a_i, later w2, later partials
  float* bufB = smem + 16 * LSTR;     // [16][LSTR] : scores_e -> a_e -> w1
  float* red  = smem + 32 * LSTR;     // [16*17]

  const float* Qi  = ws + QI_OFF;
  const float* Ki  = ws + KI_OFF;
  const float* Qe  = ws + QE_OFF;
  const float* Kb  = ws + KB_OFF;
  const float* Kin = ws + KIN_OFF;
  const float* Vb  = ws + VB_OFF;
  const float* Vin = ws + VIN_OFF;
  const float* gbv = ws + GB_OFF;
  const float* giv = ws + GI_OFF;

  const int tid  = threadIdx.x;
  const int wave = tid >> 5;
  const int lane = tid & 31;
  const int half = lane >> 4;     // WMMA lane group
  const int r    = lane & 15;     // row (A/M) or col (B/N) within tile
  const int c0   = blockIdx.x * 16;
  const float bg = *bgp;

  // ===== Phase 1: S1 = (Qi[c0:+16] @ Ki^T) * 0.25  -> bufA ==================
  v2f aq[4];
#pragma unroll
  for (int ks = 0; ks < 4; ++ks) {
    const float* q = Qi + (c0 + r) * 16 + ks * 4 + 2 * half;
    aq[ks].x = q[0];
    aq[ks].y = q[1];
  }
  for (int t8 = 0; t8 < 8; ++t8) {
    const int n0 = (wave * 8 + t8) * 16;
    v8f acc = {};
#pragma unroll
    for (int ks = 0; ks < 4; ++ks) {
      const float* kp = Ki + (n0 + r) * 16 + ks * 4 + 2 * half;
      v2f b;
      b.x = kp[0];
      b.y = kp[1];
      acc = WMMA_F32(aq[ks], b, acc);
    }
#pragma unroll
    for (int v = 0; v < 8; ++v)
      bufA[(v + 8 * half) * LSTR + n0 + r] = acc[v] * 0.25f;
  }
  __syncthreads();

  // ===== softmax -> a_i in bufA =============================================
  softmax_rows(bufA, red, tid);
  __syncthreads();

  // ===== Phase 2: scores_e = (Qe@Kb^T + a_i * Qe@Kin^T) * 0.125 -> bufB =====
  v2f aqe[16];
#pragma unroll
  for (int ks = 0; ks < 16; ++ks) {
    const float* q = Qe + (c0 + r) * 64 + ks * 4 + 2 * half;
    aqe[ks].x = q[0];
    aqe[ks].y = q[1];
  }
  for (int t8 = 0; t8 < 8; ++t8) {
    const int n0 = (wave * 8 + t8) * 16;
    v8f sb = {}, si = {};
#pragma unroll
    for (int ks = 0; ks < 16; ++ks) {
      const float* pb = Kb + (n0 + r) * 64 + ks * 4 + 2 * half;
      v2f b1;
      b1.x = pb[0];
      b1.y = pb[1];
      sb = WMMA_F32(aqe[ks], b1, sb);
      const float* pi = Kin + (n0 + r) * 64 + ks * 4 + 2 * half;
      v2f b2;
      b2.x = pi[0];
      b2.y = pi[1];
      si = WMMA_F32(aqe[ks], b2, si);
    }
#pragma unroll
    for (int v = 0; v < 8; ++v) {
      const int row = v + 8 * half;
      const float ai = bufA[row * LSTR + n0 + r];
      bufB[row * LSTR + n0 + r] = (sb[v] + ai * si[v]) * 0.125f;
    }
  }
  __syncthreads();

  // ===== softmax -> a_e in bufB =============================================
  softmax_rows(bufB, red, tid);
  __syncthreads();

  // ===== Phase 3: gate; bufB <- w1 = g*a_e ; bufA <- w2 = w1*a_i ============
  for (int e = tid; e < 16 * NN; e += 256) {
    const int row = e >> 10;
    const int col = e & 1023;
    const int idx = row * LSTR + col;
    const float ae = bufB[idx];
    const float ai = bufA[idx];
    const float z  = ae * (gbv[col] + ai * giv[col]) + bg;
    const float g  = 1.0f / (1.0f + __expf(-z));
    const float w1 = g * ae;
    bufB[idx] = w1;
    bufA[idx] = w1 * ai;
  }
  __syncthreads();

  // ===== Phase 4: out = w1 @ Vb + w2 @ Vin (K split across 8 waves) =========
  v8f acc[4] = {};
  const int kbase = wave * 128;
  for (int kt = 0; kt < 32; ++kt) {
    const int kk = kbase + kt * 4 + 2 * half;
    v2f a1, a2;
    a1.x = bufB[r * LSTR + kk];
    a1.y = bufB[r * LSTR + kk + 1];
    a2.x = bufA[r * LSTR + kk];
    a2.y = bufA[r * LSTR + kk + 1];
#pragma unroll
    for (int ct = 0; ct < 4; ++ct) {
      v2f b1, b2;
      b1.x = Vb[kk * 64 + ct * 16 + r];
      b1.y = Vb[(kk + 1) * 64 + ct * 16 + r];
      acc[ct] = WMMA_F32(a1, b1, acc[ct]);
      b2.x = Vin[kk * 64 + ct * 16 + r];
      b2.y = Vin[(kk + 1) * 64 + ct * 16 + r];
      acc[ct] = WMMA_F32(a2, b2, acc[ct]);
    }
  }
  __syncthreads();  // done reading w1/w2; reuse bufA for partials

  float* part = bufA;  // [8 waves][16][64] = 8192 floats
#pragma unroll
  for (int ct = 0; ct < 4; ++ct)
#pragma unroll
    for (int v = 0; v < 8; ++v)
      part[(wave * 16 + v + 8 * half) * 64 + ct * 16 + r] = acc[ct][v];
  __syncthreads();

  for (int e = tid; e < 16 * 64; e += 256) {
    const int row = e >> 6;
    const int col = e & 63;
    float s = 0.f;
#pragma unroll
    for (int w = 0; w < 8; ++w) s += part[(w * 16 + row) * 64 + col];
    out[(c0 + row) * 64 + col] = s;
  }
}

// ---------------------------------------------------------------------------
extern "C" void kernel_launch(void* const* d_in, const int* in_sizes, int n_in,
                              void* d_out, int out_size, void* d_ws, size_t ws_size,
                              hipStream_t stream) {
  const float* X    = (const float*)d_in[0];
  const float* Wq_i = (const float*)d_in[1];
  const float* Wk_i = (const float*)d_in[2];
  const float* Wv_i = (const float*)d_in[3];
  const float* Wq_e = (const float*)d_in[4];
  const float* Wk_e = (const float*)d_in[5];
  const float* Wv_e = (const float*)d_in[6];
  const float* wg   = (const float*)d_in[7];
  const float* bg   = (const float*)d_in[8];
  float* out = (float*)d_out;
  float* ws  = (float*)d_ws;  // needs WS_FLOATS*4 ~= 1.45 MB

  nce_precompute<<<4, 256, 0, stream>>>(X, Wq_i, Wk_i, Wv_i, Wq_e, Wk_e, Wv_e, wg, ws);

  const size_t smem_bytes = (size_t)(2 * 16 * LSTR + 16 * 17) * sizeof(float);  // ~132.7 KB
  nce_fused<<<NN / 16, 256, smem_bytes, stream>>>(ws, bg, out);
}